// MultiHeadAttention_7705171329687
// MI455X (gfx1250) — compile-verified
//
#include <hip/hip_runtime.h>
#include <math.h>

typedef __attribute__((ext_vector_type(2))) float v2f;
typedef __attribute__((ext_vector_type(8))) float v8f;

#define N_NODES  20000
#define N_EDGES  640000
#define HIDDEN   128
#define HEADS    8
#define HEAD_DIM 16
#define SCALE_F  0.25f   /* HEAD_DIM^-0.5 */
#define EPS_F    1e-8f

#define KCHUNK    32
#define KP_STRIDE 288    /* dwords per k-pair row: 2*128 + 32 pad -> kpair step = 32 banks,
                            so half-wave bank ranges are disjoint for ds_load_b64 */

// ---------------------------------------------------------------------------
// WMMA GEMM: C[M x 128] = X[M x 128] @ W[128 x 128]^T + b   (row-major all)
// Block = 64 threads (2 waves). Each wave computes a full 16x128 row stripe
// (8 fp32 WMMA accumulators), so each A fragment feeds 8 WMMAs. W is staged
// per 32-wide K chunk into LDS in PAIR-INTERLEAVED form:
//     wt[(k>>1)*KP_STRIDE + 2*n + (k&1)] = W[n][kc+k]
// so each B fragment {B[k][n], B[k+1][n]} is a single aligned ds_load_b64
// straight into an even VGPR pair (no repack movs).
// M = 20000 = 625 blocks * 2 waves * 16 rows exactly -> no guards, EXEC all 1s.
// ---------------------------------------------------------------------------
__global__ __launch_bounds__(64)
void gat_gemm_xwT_bias(const float* __restrict__ X, const float* __restrict__ W,
                       const float* __restrict__ b, float* __restrict__ C) {
    __shared__ float wt[(KCHUNK / 2) * KP_STRIDE];   // 18.4 KB

    const int tid  = threadIdx.x;       // 0..63
    const int wave = tid >> 5;          // 0..1
    const int lane = tid & 31;
    const int half = lane >> 4;         // 0 or 1
    const int l15  = lane & 15;
    const int m0   = (blockIdx.x * 2 + wave) * 16;

    // 8 accumulators, one per 16-col tile, preloaded with bias
    v8f acc[8];
#pragma unroll
    for (int nt = 0; nt < 8; ++nt) {
        const float bb = b[nt * 16 + l15];
#pragma unroll
        for (int i = 0; i < 8; ++i) acc[nt][i] = bb;
    }

    // A fragment source: X[m0+l15][k = kc+kk+2*half + {0,1}]
    const float* aptr = X + (m0 + l15) * HIDDEN + 2 * half;

#pragma unroll 1
    for (int kc = 0; kc < HIDDEN; kc += KCHUNK) {
        __syncthreads();   // protect previous chunk reads before overwrite
        // stage W chunk pair-interleaved; coalesced global reads (lane -> k)
#pragma unroll 8
        for (int i = 0; i < 64; ++i) {
            const int idx = i * 64 + tid;          // 0..4095
            const int n = idx >> 5;                // 0..127
            const int k = idx & 31;                // 0..31
            wt[(k >> 1) * KP_STRIDE + 2 * n + (k & 1)] = W[n * HIDDEN + kc + k];
        }
        __syncthreads();

        if (kc + KCHUNK < HIDDEN)                   // prefetch next A chunk
            __builtin_prefetch(aptr + kc + KCHUNK, 0, 0);

#pragma unroll
        for (int kk = 0; kk < KCHUNK; kk += 4) {
            const v2f av = *(const v2f*)(aptr + kc + kk);
            // k-pair row for this half-wave: (kk + 2*half) >> 1
            const float* bl = wt + ((kk >> 1) + half) * KP_STRIDE + 2 * l15;
#pragma unroll
            for (int nt = 0; nt < 8; ++nt) {
                const v2f bv = *(const v2f*)(bl + nt * 32);   // ds_load_b64
                acc[nt] = __builtin_amdgcn_wmma_f32_16x16x4_f32(
                    /*neg_a=*/false, av, /*neg_b=*/false, bv,
                    /*c_mod=*/(short)0, acc[nt], /*reuse_a=*/false, /*reuse_b=*/false);
            }
        }
    }

    // D layout: VGPR r -> row m0 + r + 8*half, col nt*16 + l15
    float* out = C + (m0 + 8 * half) * HIDDEN + l15;
#pragma unroll
    for (int nt = 0; nt < 8; ++nt)
#pragma unroll
        for (int r = 0; r < 8; ++r)
            out[r * HIDDEN + nt * 16] = acc[nt][r];
}

// ---------------------------------------------------------------------------
// workspace init: seg_max = -inf, seg_sum = 0, agg = 0
// ---------------------------------------------------------------------------
__global__ void gat_init_ws(float* __restrict__ smax, float* __restrict__ ssum,
                            float* __restrict__ agg) {
    const int t = blockIdx.x * blockDim.x + threadIdx.x;
    if (t < N_NODES * HEADS) {
        smax[t] = -__builtin_huge_valf();
        ssum[t] = 0.0f;
    }
    if (t < N_NODES * HIDDEN) agg[t] = 0.0f;
}

// monotone float atomic-max via signed/unsigned integer atomics
__device__ __forceinline__ void atomicMaxFloat(float* addr, float v) {
    if (v >= 0.0f) atomicMax((int*)addr, __float_as_int(v));
    else           atomicMin((unsigned int*)addr, __float_as_uint(v));
}

// ---------------------------------------------------------------------------
// per (edge, head) attention scores + segment max
// ---------------------------------------------------------------------------
__global__ void gat_edge_scores(const float* __restrict__ Q, const float* __restrict__ K,
                                const int* __restrict__ src, const int* __restrict__ dst,
                                float* __restrict__ sc, float* __restrict__ smax) {
    const int t = blockIdx.x * blockDim.x + threadIdx.x;
    if (t >= N_EDGES * HEADS) return;
    const int e = t >> 3, h = t & 7;
    const int s = src[e], d = dst[e];
    const float4* q = (const float4*)(Q + d * HIDDEN + h * HEAD_DIM);
    const float4* k = (const float4*)(K + s * HIDDEN + h * HEAD_DIM);
    float acc = 0.0f;
#pragma unroll
    for (int i = 0; i < 4; ++i) {
        float4 a = q[i], b = k[i];
        acc += a.x * b.x + a.y * b.y + a.z * b.z + a.w * b.w;
    }
    acc *= SCALE_F;
    sc[t] = acc;
    atomicMaxFloat(&smax[d * HEADS + h], acc);
}

// ---------------------------------------------------------------------------
// exp(score - seg_max[dst]) and segment sum
// ---------------------------------------------------------------------------
__global__ void gat_edge_expsum(const int* __restrict__ dst, float* __restrict__ sc,
                                const float* __restrict__ smax, float* __restrict__ ssum) {
    const int t = blockIdx.x * blockDim.x + threadIdx.x;
    if (t >= N_EDGES * HEADS) return;
    const int e = t >> 3, h = t & 7;
    const int d = dst[e];
    const float ex = expf(sc[t] - smax[d * HEADS + h]);
    sc[t] = ex;
    atomicAdd(&ssum[d * HEADS + h], ex);
}

// ---------------------------------------------------------------------------
// weighted scatter aggregation: agg[dst] += (exp/sum) * V[src]
// one thread per (edge, 4-channel group): 4-wide vector load, 4 float atomics
// ---------------------------------------------------------------------------
__global__ void gat_edge_aggregate(const int* __restrict__ src, const int* __restrict__ dst,
                                   const float* __restrict__ sc, const float* __restrict__ ssum,
                                   const float* __restrict__ V, float* __restrict__ agg) {
    const int t = blockIdx.x * blockDim.x + threadIdx.x;
    if (t >= N_EDGES * 32) return;
    const int e = t >> 5;
    const int g = t & 31;          // channel group: 4 floats each; h = g>>2
    const int h = g >> 2;
    const int s = src[e], d = dst[e];
    const float coef = sc[e * HEADS + h] / (ssum[d * HEADS + h] + EPS_F);
    const float4 v = *(const float4*)(V + s * HIDDEN + g * 4);
    float* o = agg + d * HIDDEN + g * 4;
    atomicAdd(o + 0, coef * v.x);
    atomicAdd(o + 1, coef * v.y);
    atomicAdd(o + 2, coef * v.z);
    atomicAdd(o + 3, coef * v.w);
}

// ---------------------------------------------------------------------------
extern "C" void kernel_launch(void* const* d_in, const int* in_sizes, int n_in,
                              void* d_out, int out_size, void* d_ws, size_t ws_size,
                              hipStream_t stream) {
    const float* query = (const float*)d_in[0];
    const float* key   = (const float*)d_in[1];
    const float* value = (const float*)d_in[2];
    const int*   ei    = (const int*)  d_in[3];
    const float* Wq = (const float*)d_in[4];
    const float* bq = (const float*)d_in[5];
    const float* Wk = (const float*)d_in[6];
    const float* bk = (const float*)d_in[7];
    const float* Wv = (const float*)d_in[8];
    const float* bv = (const float*)d_in[9];
    const float* Wo = (const float*)d_in[10];
    const float* bo = (const float*)d_in[11];
    float* out = (float*)d_out;

    const int* src = ei;            // edge_index[0]
    const int* dst = ei + N_EDGES;  // edge_index[1]

    // workspace partition (floats)
    float* ws   = (float*)d_ws;
    float* Q    = ws;                              // 20000*128
    float* Kp   = Q    + N_NODES * HIDDEN;         // 20000*128
    float* Vp   = Kp   + N_NODES * HIDDEN;         // 20000*128
    float* AGG  = Vp   + N_NODES * HIDDEN;         // 20000*128
    float* SC   = AGG  + N_NODES * HIDDEN;         // 640000*8
    float* SMAX = SC   + N_EDGES * HEADS;          // 20000*8
    float* SSUM = SMAX + N_NODES * HEADS;          // 20000*8

    const dim3 gemm_grid(N_NODES / 32);            // 625 blocks * 2 waves * 16 rows

    gat_init_ws<<<(N_NODES * HIDDEN + 255) / 256, 256, 0, stream>>>(SMAX, SSUM, AGG);

    gat_gemm_xwT_bias<<<gemm_grid, 64, 0, stream>>>(query, Wq, bq, Q);
    gat_gemm_xwT_bias<<<gemm_grid, 64, 0, stream>>>(key,   Wk, bk, Kp);
    gat_gemm_xwT_bias<<<gemm_grid, 64, 0, stream>>>(value, Wv, bv, Vp);

    const int eh = N_EDGES * HEADS;
    gat_edge_scores <<<(eh + 255) / 256, 256, 0, stream>>>(Q, Kp, src, dst, SC, SMAX);
    gat_edge_expsum <<<(eh + 255) / 256, 256, 0, stream>>>(dst, SC, SMAX, SSUM);

    gat_edge_aggregate<<<(N_EDGES * 32 + 255) / 256, 256, 0, stream>>>(src, dst, SC, SSUM, Vp, AGG);

    gat_gemm_xwT_bias<<<gemm_grid, 64, 0, stream>>>(AGG, Wo, bo, out);
}